// NeuralCTHMM_44590350467623
// MI455X (gfx1250) — compile-verified
//
#include <hip/hip_runtime.h>

#define GLOBAL_AS __attribute__((address_space(1)))
#define LDS_AS    __attribute__((address_space(3)))

typedef float v2f __attribute__((ext_vector_type(2)));
typedef float v8f __attribute__((ext_vector_type(8)));
typedef int   v2i __attribute__((ext_vector_type(2)));

constexpr int B_ = 1024, T_ = 8192;
constexpr int CCH   = 64;          // chunks per sequence (parallel scan split)
constexpr int LCH   = T_ / CCH;    // 128 timesteps per chunk
constexpr int BLK   = 256;         // chunk-rows per block (8 waves, wave32)
constexpr int TTILE = 16;          // timesteps staged per LDS tile
constexpr int NTILES = LCH / TTILE;

#if __has_builtin(__builtin_amdgcn_global_load_async_to_lds_b64)
#define HAVE_ASYNC 1
#else
#define HAVE_ASYNC 0
#endif

#if __has_builtin(__builtin_amdgcn_s_wait_asynccnt)
#define WAIT_ASYNC(n) __builtin_amdgcn_s_wait_asynccnt(n)
#elif HAVE_ASYNC
#define WAIT_ASYNC(n) asm volatile("s_wait_asynccnt %0" :: "n"(n) : "memory")
#else
#define WAIT_ASYNC(n) ((void)0)
#endif

__device__ __forceinline__ void stage_b64(const float2* gp, float2* lp) {
#if HAVE_ASYNC
  // builtin signature (revealed by clang diagnostics):
  //   (v2i addrspace(1)*, v2i addrspace(3)*, int imm_offset, int cpol)
  __builtin_amdgcn_global_load_async_to_lds_b64(
      (GLOBAL_AS v2i*)(GLOBAL_AS void*)(float2*)gp,
      (LDS_AS v2i*)(LDS_AS void*)lp, 0, 0);
#else
  *lp = *gp;   // fallback: sync load + LDS store
#endif
}

__device__ __forceinline__ float lse2(float a, float b) {
  float hi = fmaxf(a, b), lo = fminf(a, b);
  return hi + log1pf(__expf(lo - hi));
}

// Kernel 1: each lane computes the 2x2 log-domain chunk operator for one
// (sequence b, chunk c). Chunk 0 folds in the uniform prior (constant rows).
__global__ __launch_bounds__(BLK) void cthmm_scan_kernel(
    const float* __restrict__ seq, const float* __restrict__ means,
    const float* __restrict__ log_vars, const float* __restrict__ log_rates,
    float4* __restrict__ ws)
{
  // transposed tile: [buf][t*BLK + row] (float2 = one timestep's 2 features)
  __shared__ float2 tile[2][TTILE * BLK];   // 64 KiB

  const int l = threadIdx.x;                 // row within block = chunk lane
  const int u = blockIdx.x * BLK + l;        // global chunk id = b*CCH + c
  const int c = u & (CCH - 1);

  // tiny parameter tiles (scalar-cached loads)
  const float m00 = means[0], m01 = means[1], m10 = means[2], m11 = means[3];
  const float lv00 = log_vars[0], lv01 = log_vars[1];
  const float lv10 = log_vars[2], lv11 = log_vars[3];
  const float iv00 = __expf(-lv00), iv01 = __expf(-lv01);
  const float iv10 = __expf(-lv10), iv11 = __expf(-lv11);
  const float LOG2PI = 1.8378770664093453f;
  const float k0 = -0.5f * (2.f * LOG2PI + lv00 + lv01);
  const float k1 = -0.5f * (2.f * LOG2PI + lv10 + lv11);
  const float lt00 = -__expf(log_rates[0]);
  const float lt01 = -__expf(log_rates[1]);
  const float lt10 = -__expf(log_rates[2]);
  const float lt11 = -__expf(log_rates[3]);

  // chunk u's data is contiguous: seq viewed as [B*CCH][LCH] float2
  const float2* gbase = (const float2*)seq + (size_t)blockIdx.x * BLK * LCH;

  const int tsub = l & 15;     // timestep this lane stages
  const int rsub = l >> 4;     // row sub-index within 16-row group

  auto issue = [&](int tIdx, int buf) {
#pragma unroll
    for (int i = 0; i < 16; ++i) {
      int row = i * 16 + rsub;   // 16 rows per issue, 16 lanes cover 16 t's
      stage_b64(&gbase[(size_t)row * LCH + tIdx * TTILE + tsub],
                &tile[buf][tsub * BLK + row]);
    }
  };

  float M00 = 0.f, M01 = 0.f, M10 = 0.f, M11 = 0.f;

  issue(0, 0);
  for (int tIdx = 0; tIdx < NTILES; ++tIdx) {
    const int buf = tIdx & 1;
    if (tIdx + 1 < NTILES) { issue(tIdx + 1, buf ^ 1); WAIT_ASYNC(16); }
    else                   { WAIT_ASYNC(0); }
    __syncthreads();
#pragma unroll
    for (int tt = 0; tt < TTILE; ++tt) {
      float2 y = tile[buf][tt * BLK + l];            // conflict-free
      float d0 = y.x - m00, d1 = y.y - m01;
      float e0 = k0 - 0.5f * (d0 * d0 * iv00 + d1 * d1 * iv01);
      d0 = y.x - m10; d1 = y.y - m11;
      float e1 = k1 - 0.5f * (d0 * d0 * iv10 + d1 * d1 * iv11);
      const int tg = tIdx * TTILE + tt;
      if (tg == 0) {
        if (c == 0) {  // fold uniform prior: rows identical
          M00 = e0 - 0.69314718f; M01 = e1 - 0.69314718f; M10 = M00; M11 = M01;
        } else {       // first operator of the chunk: A_t = lt + em
          M00 = lt00 + e0; M01 = lt01 + e1; M10 = lt10 + e0; M11 = lt11 + e1;
        }
      } else {         // M <- M ∘ A_t  (log-matrix product)
        float n00 = e0 + lse2(M00 + lt00, M01 + lt10);
        float n01 = e1 + lse2(M00 + lt01, M01 + lt11);
        float n10 = e0 + lse2(M10 + lt00, M11 + lt10);
        float n11 = e1 + lse2(M10 + lt01, M11 + lt11);
        M00 = n00; M01 = n01; M10 = n10; M11 = n11;
      }
    }
    __syncthreads();
  }
  ws[u] = make_float4(M00, M01, M10, M11);
}

// Kernel 2: per sequence, compose the 64 chunk operators (lane b), then a
// WMMA-based mean reduction of the 1024 per-sequence log-likelihoods.
__global__ __launch_bounds__(1024) void cthmm_combine_kernel(
    const float4* __restrict__ ws, float* __restrict__ out)
{
  __shared__ float lls[B_];
  const int b = threadIdx.x;
  float4 m0 = ws[(size_t)b * CCH];
  float la0 = m0.x, la1 = m0.y;                 // chunk-0 rows are identical
  for (int cc = 1; cc < CCH; ++cc) {
    float4 M = ws[(size_t)b * CCH + cc];
    float n0 = lse2(la0 + M.x, la1 + M.z);
    float n1 = lse2(la0 + M.y, la1 + M.w);
    la0 = n0; la1 = n1;
  }
  lls[b] = lse2(la0, la1);
  __syncthreads();

  if (threadIdx.x < 32) {   // wave 0, EXEC all-ones: WMMA mean reduction
    const int L = threadIdx.x;
    v8f acc{};
    v2f ones; ones[0] = 1.0f; ones[1] = 1.0f;   // B = all-ones 4x16
#pragma unroll
    for (int q = 0; q < 16; ++q) {              // 16 tiles of 64 values
      const int base = q * 64;
      v2f a;   // A[16x4] layout: lanes 0-15 K=0,1 ; lanes 16-31 K=2,3
      if (L < 16) { a[0] = lls[base + L * 4 + 0]; a[1] = lls[base + L * 4 + 1]; }
      else        { a[0] = lls[base + (L - 16) * 4 + 2];
                    a[1] = lls[base + (L - 16) * 4 + 3]; }
      acc = __builtin_amdgcn_wmma_f32_16x16x4_f32(
          false, a, false, ones, (short)0, acc, false, false);
    }
    // column 0 of D: lane 0 holds M=0..7, lane 16 holds M=8..15
    float s = acc[0] + acc[1] + acc[2] + acc[3] + acc[4] + acc[5] + acc[6] + acc[7];
    float tot = __shfl(s, 0, 32) + __shfl(s, 16, 32);
    if (L == 0) out[0] = tot * (1.0f / (float)B_);
  }
}

extern "C" void kernel_launch(void* const* d_in, const int* in_sizes, int n_in,
                              void* d_out, int out_size, void* d_ws, size_t ws_size,
                              hipStream_t stream) {
  const float* seq       = (const float*)d_in[0];
  const float* means     = (const float*)d_in[1];
  const float* log_vars  = (const float*)d_in[2];
  const float* log_rates = (const float*)d_in[3];
  float4* ws  = (float4*)d_ws;     // B*CCH 2x2 log-matrices = 1 MiB
  float*  out = (float*)d_out;

  const int nblocks = (B_ * CCH) / BLK;   // 256 blocks x 256 threads
  hipLaunchKernelGGL(cthmm_scan_kernel, dim3(nblocks), dim3(BLK), 0, stream,
                     seq, means, log_vars, log_rates, ws);
  hipLaunchKernelGGL(cthmm_combine_kernel, dim3(1), dim3(B_), 0, stream,
                     ws, out);
}